// BatchedDiffPoolLayer_70214125355087
// MI455X (gfx1250) — compile-verified
//
#include <hip/hip_runtime.h>
#include <math.h>

#define NN   12288
#define EE   196608
#define INF_ 128
#define ADIM 256
#define KDIM 256      // 2*IN_FEATS
#define CPG  32
#define TS_LOG 19
#define TS   (1 << TS_LOG)

typedef __attribute__((ext_vector_type(2))) float v2f;
typedef __attribute__((ext_vector_type(8))) float v8f;

__device__ __forceinline__ v8f wmma_f32(v2f a, v2f b, v8f c) {
  // 8 args: (neg_a, A, neg_b, B, c_mod, C, reuse_a, reuse_b)
  return __builtin_amdgcn_wmma_f32_16x16x4_f32(false, a, false, b, (short)0, c,
                                               false, false);
}

// ---------------------------------------------------------------------------
// 1) agg[dst] += x[src]; deg[dst] += 1.   One wave (32 lanes) per edge.
// ---------------------------------------------------------------------------
__global__ void k_scatter_agg(const float* __restrict__ x,
                              const int* __restrict__ ei,
                              float* __restrict__ agg,
                              float* __restrict__ deg) {
  int wave = threadIdx.x >> 5;
  int lane = threadIdx.x & 31;
  int e = blockIdx.x * 8 + wave;
  if (e >= EE) return;
  int src = ei[e];
  int dst = ei[EE + e];
  const float4* xs = (const float4*)(x + (size_t)src * INF_);
  float4 v = xs[lane];                      // 32 lanes x 4 floats = 128 feats
  float* ad = agg + (size_t)dst * INF_ + lane * 4;
  atomicAdd(ad + 0, v.x);
  atomicAdd(ad + 1, v.y);
  atomicAdd(ad + 2, v.z);
  atomicAdd(ad + 3, v.w);
  if (lane == 0) atomicAdd(deg + dst, 1.0f);
}

// ---------------------------------------------------------------------------
// 2) Fused SAGE conv: h = [x || agg/deg] @ W + b, row L2-normalize,
//    optional masked-softmax epilogue (pool path).  16 rows per block,
//    blockDim == OW (128 -> 4 waves, 256 -> 8 waves), 2 N-tiles per wave.
// ---------------------------------------------------------------------------
template <int OW, bool SOFTMAX>
__global__ void k_sage(const float* __restrict__ x,
                       const float* __restrict__ agg,
                       const float* __restrict__ deg,
                       const float* __restrict__ W,
                       const float* __restrict__ bias,
                       const int* __restrict__ batch,
                       float* __restrict__ out) {
  constexpr int AP = KDIM + 4;   // LDS pitch (bank-conflict pad)
  constexpr int HP = OW + 8;
  __shared__ float As[16 * AP];
  __shared__ float Hs[16 * HP];
  __shared__ float invs[16];
  __shared__ float rhos[16];
  __shared__ int   gbs[16];

  const int tid = threadIdx.x;
  const int row0 = blockIdx.x * 16;

  // stage A tile = concat(x, agg/deg) rows [row0, row0+16)
  for (int i = tid; i < 16 * KDIM; i += OW) {
    int m = i >> 8;
    int k = i & 255;
    int row = row0 + m;
    float v;
    if (k < INF_) v = x[(size_t)row * INF_ + k];
    else          v = agg[(size_t)row * INF_ + (k - INF_)] / fmaxf(deg[row], 1.0f);
    As[m * AP + k] = v;
  }
  __syncthreads();

  const int wave = tid >> 5, lane = tid & 31;
  const int half = lane >> 4, mn = lane & 15;

  for (int t = 0; t < 2; ++t) {
    const int cn = (wave * 2 + t) * 16 + mn;
    v8f acc = {};
    for (int k0 = 0; k0 < KDIM; k0 += 4) {
      const int kk = k0 + 2 * half;
      v2f a, b;
      a.x = As[mn * AP + kk];
      a.y = As[mn * AP + kk + 1];
      b.x = W[(size_t)kk * OW + cn];
      b.y = W[(size_t)(kk + 1) * OW + cn];
      acc = wmma_f32(a, b, acc);
    }
    const float bv = bias[cn];
#pragma unroll
    for (int r = 0; r < 8; ++r) {
      const int M = r + 8 * half;           // C layout: N=lane%16, M=r+8*(lane/16)
      Hs[M * HP + cn] = acc[r] + bv;
    }
  }
  __syncthreads();

  if (tid < 16) {
    float ss = 0.0f;
    for (int c = 0; c < OW; ++c) { float v = Hs[tid * HP + c]; ss += v * v; }
    const float inv = 1.0f / fmaxf(sqrtf(ss), 1e-12f);
    invs[tid] = inv;
    if constexpr (SOFTMAX) {
      const int gb = batch[row0 + tid] * CPG;
      gbs[tid] = gb;
      float Sb = 0.0f;
      for (int c = gb; c < gb + CPG; ++c) Sb += expf(Hs[tid * HP + c] * inv);
      const float D = Sb + (float)(OW - CPG);     // off-block exp(0) mass
      rhos[tid] = 1.0f / (Sb + 1e-13f * D);
    }
  }
  __syncthreads();

  for (int i = tid; i < 16 * OW; i += OW) {
    const int m = i / OW, c = i % OW;
    float v = Hs[m * HP + c] * invs[m];
    if constexpr (SOFTMAX) {
      const int gb = gbs[m];
      v = (c >= gb && c < gb + CPG) ? expf(v) * rhos[m] : 0.0f;
    }
    out[(size_t)(row0 + m) * OW + c] = v;
  }
}

// ---------------------------------------------------------------------------
// 3) Ar[src] += result[dst]  and  per-block partials of <A, S S^T>.
//    One wave per edge; deterministic block-level partial sums.
// ---------------------------------------------------------------------------
__global__ void k_scatter_Ar(const float* __restrict__ result,
                             const int* __restrict__ ei,
                             float* __restrict__ Ar,
                             float* __restrict__ dotp) {
  __shared__ float wsum[8];
  const int wave = threadIdx.x >> 5, lane = threadIdx.x & 31;
  const int e = blockIdx.x * 8 + wave;
  float p = 0.0f;
  if (e < EE) {
    const int src = ei[e], dst = ei[EE + e];
    const float4* rd = (const float4*)(result + (size_t)dst * ADIM);
    const float4* rs = (const float4*)(result + (size_t)src * ADIM);
    const float4 d0 = rd[lane * 2], d1 = rd[lane * 2 + 1];
    const float4 s0 = rs[lane * 2], s1 = rs[lane * 2 + 1];
    float* a = Ar + (size_t)src * ADIM + lane * 8;
    atomicAdd(a + 0, d0.x); atomicAdd(a + 1, d0.y);
    atomicAdd(a + 2, d0.z); atomicAdd(a + 3, d0.w);
    atomicAdd(a + 4, d1.x); atomicAdd(a + 5, d1.y);
    atomicAdd(a + 6, d1.z); atomicAdd(a + 7, d1.w);
    p = d0.x * s0.x + d0.y * s0.y + d0.z * s0.z + d0.w * s0.w +
        d1.x * s1.x + d1.y * s1.y + d1.z * s1.z + d1.w * s1.w;
  }
  for (int off = 16; off >= 1; off >>= 1) p += __shfl_xor(p, off, 32);
  if (lane == 0) wsum[wave] = p;
  __syncthreads();
  if (threadIdx.x == 0) {
    float s = 0.0f;
    for (int w = 0; w < 8; ++w) s += wsum[w];
    dotp[blockIdx.x] = s;
  }
}

// ---------------------------------------------------------------------------
// 4) C[ADIM, ncols] = A^T @ B  with A = result [NN, ADIM], K = NN.
//    One 16x16 output tile per block; 8 waves split K, LDS cross-wave reduce.
// ---------------------------------------------------------------------------
__global__ void k_atb(const float* __restrict__ A,
                      const float* __restrict__ B,
                      float* __restrict__ C,
                      int ncols, int tilesN) {
  __shared__ float red[8 * 32 * 8];
  const int tm = blockIdx.x / tilesN;
  const int tn = blockIdx.x % tilesN;
  const int tid = threadIdx.x;
  const int wave = tid >> 5, lane = tid & 31;
  const int half = lane >> 4, mn = lane & 15;
  const int cm = tm * 16 + mn;
  const int cn = tn * 16 + mn;

  v8f acc = {};
  for (int k = wave * 4; k < NN; k += 32) {
    const int kk = k + 2 * half;
    v2f a, b;
    a.x = A[(size_t)kk * ADIM + cm];        // A^T[m][k] = result[k][m]
    a.y = A[(size_t)(kk + 1) * ADIM + cm];
    b.x = B[(size_t)kk * ncols + cn];
    b.y = B[(size_t)(kk + 1) * ncols + cn];
    acc = wmma_f32(a, b, acc);
  }
#pragma unroll
  for (int r = 0; r < 8; ++r) red[(wave * 32 + lane) * 8 + r] = acc[r];
  __syncthreads();

  const int lane_id = tid & 31;
  const int r = tid >> 5;
  float s = 0.0f;
#pragma unroll
  for (int w = 0; w < 8; ++w) s += red[(w * 32 + lane_id) * 8 + r];
  const int M = r + 8 * (lane_id >> 4);
  const int Ncol = lane_id & 15;
  C[(size_t)(tm * 16 + M) * ncols + tn * 16 + Ncol] = s;
}

// ---------------------------------------------------------------------------
// 5) Open-addressing hash of edge pairs -> exact Sum A_ij^2 (duplicates).
// ---------------------------------------------------------------------------
__global__ void k_hash(const int* __restrict__ ei,
                       int* __restrict__ keys, int* __restrict__ counts) {
  const int e = blockIdx.x * blockDim.x + threadIdx.x;
  if (e >= EE) return;
  const int key = ei[e] * NN + ei[EE + e];          // < 2^31
  unsigned slot = (((unsigned)key) * 2654435761u) & (TS - 1);
  for (;;) {
    const int old = atomicCAS(&keys[slot], -1, key);
    if (old == -1 || old == key) { atomicAdd(&counts[slot], 1); break; }
    slot = (slot + 1) & (TS - 1);
  }
}

// ---------------------------------------------------------------------------
// 6) Finalize scalars: adj_loss and (constant) entropy.
// ---------------------------------------------------------------------------
__global__ void k_final(const int* __restrict__ counts,
                        const float* __restrict__ G,
                        const float* __restrict__ dotp, int ndot,
                        float* __restrict__ out_loss,
                        float* __restrict__ out_ent) {
  __shared__ float     sf[256];
  __shared__ long long si[256];
  const int tid = threadIdx.x;

  long long aa = 0;
  for (int i = tid; i < TS; i += 256) { long long c = counts[i]; aa += c * c; }
  float sg = 0.0f;
  for (int i = tid; i < ADIM * ADIM; i += 256) { float v = G[i]; sg += v * v; }
  float sd = 0.0f;
  for (int i = tid; i < ndot; i += 256) sd += dotp[i];

  si[tid] = aa; sf[tid] = sg;
  __syncthreads();
  for (int s = 128; s > 0; s >>= 1) {
    if (tid < s) { si[tid] += si[tid + s]; sf[tid] += sf[tid + s]; }
    __syncthreads();
  }
  float aaf = 0.0f, sgf = 0.0f;
  if (tid == 0) { aaf = (float)si[0]; sgf = sf[0]; }
  __syncthreads();
  sf[tid] = sd;
  __syncthreads();
  for (int s = 128; s > 0; s >>= 1) {
    if (tid < s) sf[tid] += sf[tid + s];
    __syncthreads();
  }
  if (tid == 0) {
    const float sdf = sf[0];
    const float v = aaf - 2.0f * sdf + sgf;   // ||A - S S^T||_F^2
    *out_loss = sqrtf(fmaxf(v, 0.0f)) / ((float)NN * (float)NN);
    *out_ent  = (float)NN * logf((float)CPG); // mask rows are uniform over 32
  }
}

// ---------------------------------------------------------------------------
extern "C" void kernel_launch(void* const* d_in, const int* in_sizes, int n_in,
                              void* d_out, int out_size, void* d_ws,
                              size_t ws_size, hipStream_t stream) {
  (void)in_sizes; (void)n_in; (void)out_size; (void)ws_size;
  const float* x      = (const float*)d_in[0];
  const int*   ei     = (const int*)d_in[1];
  const int*   batch  = (const int*)d_in[2];
  const float* W_embd = (const float*)d_in[3];
  const float* b_embd = (const float*)d_in[4];
  const float* W_pool = (const float*)d_in[5];
  const float* b_pool = (const float*)d_in[6];
  float* out = (float*)d_out;   // [adj_new 256x256 | h 256x128 | loss | entropy]

  char* ws = (char*)d_ws;
  size_t off = 0;
  auto alloc = [&](size_t bytes) -> void* {
    void* p = ws + off;
    off += (bytes + 255) & ~(size_t)255;
    return p;
  };
  float* agg    = (float*)alloc((size_t)NN * INF_ * 4);
  float* deg    = (float*)alloc((size_t)NN * 4);
  float* embed  = (float*)alloc((size_t)NN * INF_ * 4);
  float* result = (float*)alloc((size_t)NN * ADIM * 4);
  float* Ar     = (float*)alloc((size_t)NN * ADIM * 4);
  float* Gm     = (float*)alloc((size_t)ADIM * ADIM * 4);
  int*   keys   = (int*)alloc((size_t)TS * 4);
  int*   counts = (int*)alloc((size_t)TS * 4);
  float* dotp   = (float*)alloc((size_t)(EE / 8) * 4);

  hipMemsetAsync(agg,    0,    (size_t)NN * INF_ * 4, stream);
  hipMemsetAsync(deg,    0,    (size_t)NN * 4,        stream);
  hipMemsetAsync(Ar,     0,    (size_t)NN * ADIM * 4, stream);
  hipMemsetAsync(counts, 0,    (size_t)TS * 4,        stream);
  hipMemsetAsync(keys,   0xFF, (size_t)TS * 4,        stream);

  k_scatter_agg<<<EE / 8, 256, 0, stream>>>(x, ei, agg, deg);

  k_sage<128, false><<<NN / 16, 128, 0, stream>>>(x, agg, deg, W_embd, b_embd,
                                                  batch, embed);
  k_sage<256, true ><<<NN / 16, 256, 0, stream>>>(x, agg, deg, W_pool, b_pool,
                                                  batch, result);

  k_scatter_Ar<<<EE / 8, 256, 0, stream>>>(result, ei, Ar, dotp);

  // h = result^T @ embed   -> out + 65536  (256 x 128)
  k_atb<<<16 * 8,  256, 0, stream>>>(result, embed, out + ADIM * ADIM, 128, 8);
  // adj_new = result^T @ (A @ result)  -> out     (256 x 256)
  k_atb<<<16 * 16, 256, 0, stream>>>(result, Ar, out, 256, 16);
  // G = result^T @ result  (for ||S S^T||_F^2 = ||G||_F^2)
  k_atb<<<16 * 16, 256, 0, stream>>>(result, result, Gm, 256, 16);

  k_hash<<<EE / 256, 256, 0, stream>>>(ei, keys, counts);

  k_final<<<1, 256, 0, stream>>>(counts, Gm, dotp, EE / 8,
                                 out + ADIM * ADIM + ADIM * INF_,
                                 out + ADIM * ADIM + ADIM * INF_ + 1);
}